// OrbitalStabilityEvaluator_29910152249677
// MI455X (gfx1250) — compile-verified
//
#include <hip/hip_runtime.h>
#include <hip/hip_bf16.h>
#include <math.h>

// Problem constants from the reference setup_inputs()
#define BB 4
#define TT 256
#define NN 64
#define FF 129          // T/2 + 1 rfft bins
#define DT_C 0.01f
#define VCLAMP_C 10.0f

typedef __attribute__((ext_vector_type(2))) float v2f;
typedef __attribute__((ext_vector_type(8))) float v8f;

// ---------------- block reduction helpers (LDS) ----------------
__device__ __forceinline__ float blk_sum(float v, float* red) {
    int tid = threadIdx.x;
    red[tid] = v; __syncthreads();
    for (int s = blockDim.x >> 1; s > 0; s >>= 1) {
        if (tid < s) red[tid] += red[tid + s];
        __syncthreads();
    }
    float r = red[0]; __syncthreads();
    return r;
}

__device__ __forceinline__ float blk_max(float v, float* red) {
    int tid = threadIdx.x;
    red[tid] = v; __syncthreads();
    for (int s = blockDim.x >> 1; s > 0; s >>= 1) {
        if (tid < s) red[tid] = fmaxf(red[tid], red[tid + s]);
        __syncthreads();
    }
    float r = red[0]; __syncthreads();
    return r;
}

// ---------------- 1) per-(b,t) total energy: KE + all-pairs PE ----------------
__global__ void k_energy(const float* __restrict__ traj, const float* __restrict__ masses,
                         const float* __restrict__ mask, const float* __restrict__ G_ref,
                         float* __restrict__ energies) {
    __shared__ float px[NN], py[NN], pz[NN], sm[NN], sw[NN];
    __shared__ float red[256];
    int bt = blockIdx.x;
    int b = bt / TT, t = bt % TT;
    int tid = threadIdx.x;
    const float* base = traj + (size_t)(b * TT + t) * NN * 3;
    if (tid < NN) {
        px[tid] = base[tid * 3 + 0];
        py[tid] = base[tid * 3 + 1];
        pz[tid] = base[tid * 3 + 2];
        sm[tid] = masses[b * NN + tid];
        sw[tid] = mask[b * NN + tid];
    }
    __syncthreads();

    float G = expf(G_ref[0]);
    G = fminf(fmaxf(G, 0.01f), 10.0f);

    float local = 0.0f;
    // kinetic energy (one particle per thread, threads 0..63)
    if (tid < NN) {
        int ta = (t < TT - 1) ? t : TT - 2;          // vel[T-1] replicates d[T-2]
        const float* pa = traj + ((size_t)(b * TT + ta) * NN + tid) * 3;
        const float* pb = pa + (size_t)NN * 3;       // tb = ta + 1
        float vx = (pb[0] - pa[0]) * (1.0f / DT_C);
        float vy = (pb[1] - pa[1]) * (1.0f / DT_C);
        float vz = (pb[2] - pa[2]) * (1.0f / DT_C);
        vx = fminf(fmaxf(vx, -VCLAMP_C), VCLAMP_C);
        vy = fminf(fmaxf(vy, -VCLAMP_C), VCLAMP_C);
        vz = fminf(fmaxf(vz, -VCLAMP_C), VCLAMP_C);
        float vsq = fminf(vx * vx + vy * vy + vz * vz, 100.0f);
        local += 0.5f * sm[tid] * vsq * sw[tid];
    }
    // potential energy: 0.5 * sum_{i!=j} -G m_i m_j / dist
    for (int p = tid; p < NN * NN; p += blockDim.x) {
        int i = p >> 6, j = p & 63;
        if (i != j) {
            float dx = px[j] - px[i];
            float dy = py[j] - py[i];
            float dz = pz[j] - pz[i];
            float dist = fmaxf(sqrtf(dx * dx + dy * dy + dz * dz), 1e-4f);
            local += -0.5f * G * sm[i] * sm[j] * sw[i] * sw[j] / dist;
        }
    }
    float E = blk_sum(local, red);
    if (tid == 0) energies[bt] = fminf(fmaxf(E, -1e4f), 1e4f);
}

// ---------------- 2) energy-conservation score (unbiased variance over T) -----
__global__ void k_energy_score(const float* __restrict__ energies, float* __restrict__ s_e) {
    __shared__ float red[256];
    int b = blockIdx.x, tid = threadIdx.x;
    float e = energies[b * TT + tid];
    float mean = blk_sum(e, red) * (1.0f / TT);
    float d = e - mean;
    float var = blk_sum(d * d, red) * (1.0f / (TT - 1));
    if (tid == 0) {
        float e0 = fabsf(energies[b * TT]) + 1e-6f;
        float ratio = fminf(var / (e0 * e0), 1e4f);
        s_e[b] = 1.0f / (1.0f + ratio);
    }
}

// ---------------- 3) centered radii rc[(b,n), t] ------------------------------
__global__ void k_rc(const float* __restrict__ traj, float* __restrict__ rc) {
    __shared__ float red[256];
    int row = blockIdx.x;               // b*NN + n
    int b = row / NN, n = row % NN;
    int t = threadIdx.x;
    const float* p = traj + ((size_t)(b * TT + t) * NN + n) * 3;
    float r = sqrtf(p[0] * p[0] + p[1] * p[1] + p[2] * p[2]);
    float mean = blk_sum(r, red) * (1.0f / TT);
    rc[row * TT + t] = r - mean;
}

// ---------------- 4) rfft magnitudes via V_WMMA_F32_16X16X4_F32 ---------------
// D(real) = rc(256x256) * cos-basis(256x144); D(imag) with sin-basis.
// One wave per 16x16 output tile; 64 K-steps of K=4 per WMMA.
// Trig is hoisted into a 256-entry LDS table (angles are p*2pi/256, p integer),
// so the inner loop is: 1 rc load + 4 ds_load + 2 WMMA, no divergence.
__global__ void __launch_bounds__(32) k_dft(const float* __restrict__ rc,
                                            float* __restrict__ mags) {
    __shared__ float ct[TT];
    __shared__ float st[TT];
    int lane = threadIdx.x;
    const float ang_unit = 6.283185307179586f / (float)TT;
    // build exact basis table once per wave (8 entries/lane, args < 2pi)
    for (int i = lane; i < TT; i += 32) {
        float a = (float)i * ang_unit;
        ct[i] = cosf(a);
        st[i] = sinf(a);
    }
    __syncthreads();

    int tile = blockIdx.x;              // 0..143
    int mt = tile & 15;                 // row tile (16 rows each, 256 rows total)
    int ft = tile >> 4;                 // freq tile (9 tiles cover 0..143 >= 129)
    int lr = lane & 15;
    bool hi = lane >= 16;
    int arow = mt * 16 + lr;            // A-matrix row (both half-waves: M = lr)
    int fcol = ft * 16 + lr;            // B/C-matrix column (frequency)
    const float* aptr = rc + arow * TT;

    v8f acc_re = {0.f, 0.f, 0.f, 0.f, 0.f, 0.f, 0.f, 0.f};
    v8f acc_im = {0.f, 0.f, 0.f, 0.f, 0.f, 0.f, 0.f, 0.f};

    for (int kk = 0; kk < TT / 4; ++kk) {
        int kb = kk * 4 + (hi ? 2 : 0); // lanes 0-15 hold K {0,1}; lanes 16-31 hold K {2,3}
        v2f a;
        a.x = aptr[kb];
        a.y = aptr[kb + 1];
        // exact integer phase: (f*t) mod T indexes the LDS basis table
        int p0 = (fcol * kb) & (TT - 1);
        int p1 = (fcol * (kb + 1)) & (TT - 1);
        v2f bc, bs;
        bc.x = ct[p0]; bc.y = ct[p1];
        bs.x = st[p0]; bs.y = st[p1];
        acc_re = __builtin_amdgcn_wmma_f32_16x16x4_f32(
            false, a, false, bc, (short)0, acc_re, false, false);
        acc_im = __builtin_amdgcn_wmma_f32_16x16x4_f32(
            false, a, false, bs, (short)0, acc_im, false, false);
    }

    if (fcol < FF) {
        // C/D layout: VGPR v holds row mt*16 + v (lanes 0-15) or + v + 8 (lanes 16-31)
        #pragma unroll
        for (int v = 0; v < 8; ++v) {
            int row = mt * 16 + v + (hi ? 8 : 0);
            float re = acc_re[v];
            float im = acc_im[v];
            float mag = (fcol == 0) ? 0.0f : sqrtf(re * re + im * im);
            mags[row * FF + fcol] = mag;
        }
    }
}

// ---------------- 5) softmax over freqs -> dominant frequency index -----------
__global__ void k_dom(const float* __restrict__ mags, const float* __restrict__ mask,
                      const float* __restrict__ log_temp, float* __restrict__ dom) {
    __shared__ float red[256];
    int row = blockIdx.x, tid = threadIdx.x;
    float temp = fminf(fmaxf(expf(log_temp[0]), 0.01f), 1.0f) + 1e-8f;
    float inv = 1.0f / temp;

    float mv = -1e30f;
    for (int f = tid; f < FF; f += blockDim.x)
        mv = fmaxf(mv, mags[row * FF + f] * inv);
    float mx = blk_max(mv, red);

    float se = 0.0f, sf = 0.0f;
    for (int f = tid; f < FF; f += blockDim.x) {
        float w = expf(mags[row * FF + f] * inv - mx);
        se += w;
        sf += w * (float)f;
    }
    float Z = blk_sum(se, red);
    float num = blk_sum(sf, red);
    if (tid == 0) {
        float d = num / Z;
        float mk = mask[row];
        dom[row] = d * mk + (1.0f - mk);   // masked particles -> 1.0
    }
}

// ---------------- 6) resonance score over all pairs ---------------------------
__global__ void k_res(const float* __restrict__ dom, const float* __restrict__ mask,
                      const float* __restrict__ log_sigma, float* __restrict__ s_r) {
    __shared__ float sd[NN], sw[NN];
    __shared__ float red[256];
    int b = blockIdx.x, tid = threadIdx.x;
    if (tid < NN) {
        sd[tid] = dom[b * NN + tid];
        sw[tid] = mask[b * NN + tid];
    }
    __syncthreads();
    float sigma = fminf(fmaxf(expf(log_sigma[0]), 0.05f), 2.0f);
    float inv_s2 = 1.0f / (sigma * sigma);

    float accR = 0.0f, accP = 0.0f;
    for (int p = tid; p < NN * NN; p += blockDim.x) {
        int i = p >> 6, j = p & 63;
        if (i == j) continue;
        float pm = sw[i] * sw[j];
        float ratio = sd[i] / (sd[j] + 1e-8f);
        // soft-min over 36 simple ratios via logsumexp(-dist/0.1)
        float mx = -1e30f;
        #pragma unroll
        for (int pp = 1; pp <= 6; ++pp) {
            #pragma unroll
            for (int qq = 1; qq <= 6; ++qq) {
                float s = -fabsf(ratio - (float)pp / (float)qq) * 10.0f;
                mx = fmaxf(mx, s);
            }
        }
        float sum = 0.0f;
        #pragma unroll
        for (int pp = 1; pp <= 6; ++pp) {
            #pragma unroll
            for (int qq = 1; qq <= 6; ++qq) {
                float s = -fabsf(ratio - (float)pp / (float)qq) * 10.0f;
                sum += expf(s - mx);
            }
        }
        float soft_min = -(mx + logf(sum));
        float res = expf(-soft_min * soft_min * inv_s2);
        accR += res * pm;
        accP += pm;
    }
    float R = blk_sum(accR, red);
    float P = blk_sum(accP, red);
    if (tid == 0) s_r[b] = R / fmaxf(P, 1.0f);
}

// ---------------- 7) compactness score ----------------------------------------
__global__ void k_comp(const float* __restrict__ traj, const float* __restrict__ mask,
                       float* __restrict__ s_c) {
    __shared__ float red[64];
    int b = blockIdx.x, n = threadIdx.x;
    const float* p0 = traj + ((size_t)(b * TT + 0) * NN + n) * 3;
    const float* pT = traj + ((size_t)(b * TT + (TT - 1)) * NN + n) * 3;
    float r0 = fmaxf(sqrtf(p0[0] * p0[0] + p0[1] * p0[1] + p0[2] * p0[2]), 1e-4f);
    float rT = sqrtf(pT[0] * pT[0] + pT[1] * pT[1] + pT[2] * pT[2]);
    float mk = mask[b * NN + n];
    float ex = rT / r0 - 1.0f;
    ex = ex * ex * mk;
    float m = blk_max(ex * mk, red);
    if (n == 0) s_c[b] = 1.0f / (1.0f + m);
}

// ---------------- 8) combine + sigmoid ----------------------------------------
__global__ void k_final(const float* __restrict__ s_e, const float* __restrict__ s_r,
                        const float* __restrict__ s_c, const float* __restrict__ alpha,
                        const float* __restrict__ beta, const float* __restrict__ gamm,
                        const float* __restrict__ bias, float* __restrict__ out) {
    int b = threadIdx.x;
    if (b < BB) {
        float z = alpha[0] * s_e[b] + beta[0] * s_r[b] + gamm[0] * s_c[b] + bias[0];
        out[b] = 1.0f / (1.0f + expf(-z));
    }
}

extern "C" void kernel_launch(void* const* d_in, const int* in_sizes, int n_in,
                              void* d_out, int out_size, void* d_ws, size_t ws_size,
                              hipStream_t stream) {
    const float* traj      = (const float*)d_in[0];
    const float* masses    = (const float*)d_in[1];
    const float* mask      = (const float*)d_in[2];
    const float* G_ref     = (const float*)d_in[3];
    const float* alpha     = (const float*)d_in[4];
    const float* beta      = (const float*)d_in[5];
    const float* gamm      = (const float*)d_in[6];
    const float* bias      = (const float*)d_in[7];
    const float* log_temp  = (const float*)d_in[8];
    const float* log_sigma = (const float*)d_in[9];
    float* out = (float*)d_out;
    float* ws  = (float*)d_ws;

    float* energies = ws;                         // B*T        = 1024
    float* rc       = energies + BB * TT;         // B*N*T      = 65536
    float* mags     = rc + BB * NN * TT;          // B*N*F      = 33024
    float* dom      = mags + BB * NN * FF;        // B*N        = 256
    float* s_e      = dom + BB * NN;              // B
    float* s_r      = s_e + BB;                   // B
    float* s_c      = s_r + BB;                   // B   (total ~390 KB << ws)

    k_energy<<<dim3(BB * TT), dim3(256), 0, stream>>>(traj, masses, mask, G_ref, energies);
    k_energy_score<<<dim3(BB), dim3(256), 0, stream>>>(energies, s_e);
    k_rc<<<dim3(BB * NN), dim3(256), 0, stream>>>(traj, rc);
    k_dft<<<dim3(16 * 9), dim3(32), 0, stream>>>(rc, mags);
    k_dom<<<dim3(BB * NN), dim3(256), 0, stream>>>(mags, mask, log_temp, dom);
    k_res<<<dim3(BB), dim3(256), 0, stream>>>(dom, mask, log_sigma, s_r);
    k_comp<<<dim3(BB), dim3(64), 0, stream>>>(traj, mask, s_c);
    k_final<<<dim3(1), dim3(BB), 0, stream>>>(s_e, s_r, s_c, alpha, beta, gamm, bias, out);
}